// LatentSDE_68719477545
// MI455X (gfx1250) — compile-verified
//
#include <hip/hip_runtime.h>
#include <hip/hip_bf16.h>

// Problem sizes (fixed by the reference).
#define B_N   8192
#define T_N   128
#define H_N   64
#define TB_N  (T_N * B_N)            // 1,048,576
#define NBLK_PW ((TB_N + 255) / 256) // 4096 blocks for pointwise kernels

typedef __attribute__((ext_vector_type(2))) float v2f;
typedef __attribute__((ext_vector_type(8))) float v8f;

// Fast stable softplus: max(x,0) + log(1 + exp(-|x|)).
// For e = exp(-|x|) <= 1 the naive log(1+e) only loses relative accuracy when
// e < ~1e-7, where the max(x,0) ~ |x| >= 16 term dominates the sum, so the
// overall relative error stays ~1e-7.  Uses v_exp_f32 / v_log_f32 directly.
__device__ __forceinline__ float softplus_f(float x) {
    float e = __expf(-fabsf(x));
    return fmaxf(x, 0.f) + __logf(1.f + e);
}

// D = A(16x4, f32) * B(4x16, f32) + C, C = per-column bias splat (bias is
// constant per lane in the C/D layout), full f32 precision.
__device__ __forceinline__ v8f wmma_16x16x4(v2f a, v2f b, float cbias) {
    v8f c = {cbias, cbias, cbias, cbias, cbias, cbias, cbias, cbias};
    return __builtin_amdgcn_wmma_f32_16x16x4_f32(
        /*neg_a=*/false, a, /*neg_b=*/false, b,
        /*c_mod=*/(short)0, c, /*reuse_a=*/false, /*reuse_b=*/false);
}

// p[r] holds, in C/D layout, per-lane partial dot-products:
//   lanes 0-15 : (batch r,   hidden-col lane)    partial
//   lanes 16-31: (batch r+8, hidden-col lane-16) partial
// Sum each 16-lane half and deposit result for batch m into lane m (m<16).
__device__ __forceinline__ float rowsum_select(const float p[8], int lane) {
    float out = 0.f;
#pragma unroll
    for (int r = 0; r < 8; ++r) {
        float v = p[r];
        v += __shfl_xor(v, 1, 32);
        v += __shfl_xor(v, 2, 32);
        v += __shfl_xor(v, 4, 32);
        v += __shfl_xor(v, 8, 32);           // halves now hold their half-sums
        float vh = __shfl_xor(v, 16, 32);    // bring batch r+8 sum into low lanes
        out = (lane == r)     ? v  : out;
        out = (lane == r + 8) ? vh : out;
    }
    return out;
}

// ---------------------------------------------------------------------------
// Kernel 1: encoder, pointwise over (t,b). ctx[t,b] = mlp(xs[t,b]).
// (flip(encoder(flip(xs))) == encoder(xs) since the encoder is pointwise.)
// ---------------------------------------------------------------------------
__global__ __launch_bounds__(256) void encoder_kernel(
    const float* __restrict__ xs,
    const float* __restrict__ ew1, const float* __restrict__ eb1,
    const float* __restrict__ ew2, const float* __restrict__ eb2,
    float* __restrict__ ctx)
{
    int i = blockIdx.x * 256 + threadIdx.x;
    if (i >= TB_N) return;
    float x = xs[i];
    float acc = eb2[0];
#pragma unroll 8
    for (int j = 0; j < H_N; ++j)
        acc += ew2[j] * softplus_f(fmaf(x, ew1[j], eb1[j]));
    ctx[i] = acc;
}

// ---------------------------------------------------------------------------
// Kernel 2: sequential SDE scan. One wave (32 lanes) advances 16 chains.
// First layers of f/h/g nets are V_WMMA_F32_16X16X4_F32 (K=2/1 padded to 4,
// bias folded into the C operand); second layers (64 -> 1) are per-lane
// partials + in-wave ds_swizzle butterfly.
// ---------------------------------------------------------------------------
__global__ __launch_bounds__(32) void sde_scan_kernel(
    const float* __restrict__ ts,
    const float* __restrict__ eps0,
    const float* __restrict__ dW,
    const float* __restrict__ qw,  const float* __restrict__ qb,
    const float* __restrict__ fw1, const float* __restrict__ fb1,
    const float* __restrict__ fw2, const float* __restrict__ fb2,
    const float* __restrict__ hw1, const float* __restrict__ hb1,
    const float* __restrict__ hw2, const float* __restrict__ hb2,
    const float* __restrict__ gw1, const float* __restrict__ gb1,
    const float* __restrict__ gw2, const float* __restrict__ gb2,
    const float* __restrict__ pz0_mean, const float* __restrict__ pz0_logstd,
    const float* __restrict__ ctx,
    float* __restrict__ zs,
    float* __restrict__ csums)
{
    const int  lane = threadIdx.x;
    const int  n16  = lane & 15;
    const bool low  = lane < 16;
    const int  base = blockIdx.x * 16;   // first chain of this wave

    // B-matrix (4x16) layout: v0 = {row K0 | row K2}, v1 = {row K1 | row K3}.
    // Rows K2/K3 are zero padding (our K is 1 or 2).
    float fB0[4], fB1[4], hB0[4], gB0[4];
    float fb1v[4], hb1v[4], gb1v[4];     // biases per hidden col (C-layout lanes)
    float fw2v[4], hw2v[4], gw2v[4];     // 2nd-layer weights per hidden col
#pragma unroll
    for (int blk = 0; blk < 4; ++blk) {
        int n = blk * 16 + n16;
        fB0[blk] = low ? fw1[n]       : 0.f;   // fw1 row 0 (z)
        fB1[blk] = low ? fw1[H_N + n] : 0.f;   // fw1 row 1 (c)
        hB0[blk] = low ? hw1[n] : 0.f;
        gB0[blk] = low ? gw1[n] : 0.f;
        fb1v[blk] = fb1[n]; hb1v[blk] = hb1[n]; gb1v[blk] = gb1[n];
        fw2v[blk] = fw2[n]; hw2v[blk] = hw2[n]; gw2v[blk] = gw2[n];
    }
    const float fb2s = fb2[0], hb2s = hb2[0], gb2s = gb2[0];
    const float qw0 = qw[0], qw1s = qw[1], qb0 = qb[0], qb1 = qb[1];
    const float pm  = pz0_mean[0], pls = pz0_logstd[0];

    // q(z0) from ctx[0, b]; KL(q0 || p0)
    float c0  = ctx[base + n16];
    float ep  = eps0[base + n16];
    float qm  = fmaf(c0, qw0, qb0);
    float qls = fmaf(c0, qw1s, qb1);
    float z   = qm + expf(qls) * ep;
    float dq  = qm - pm;
    float kl  = (pls - qls) + (expf(2.f * qls) + dq * dq) / (2.f * expf(2.f * pls)) - 0.5f;
    float acc = low ? kl : 0.f;
    if (!low) z = 0.f;
    if (low) zs[base + lane] = z;

    for (int k = 0; k < T_N - 1; ++k) {
        float dt  = ts[k + 1] - ts[k];
        float sdt = sqrtf(dt);
        // searchsorted(ts, ts[k], 'right') == k+1 for strictly increasing ts
        float cv = ctx[(k + 1) * B_N + base + n16];

        // A (16x4) layout: lanes 0-15 -> v0=K0, v1=K1 ; lanes 16-31 -> K2,K3 = 0
        v2f Af; Af.x = low ? z : 0.f; Af.y = low ? cv : 0.f;   // f-net: [z, c]
        v2f Ah; Ah.x = Af.x; Ah.y = 0.f;                        // h/g nets: [z]

        float pf[8], ph[8], pg[8];
#pragma unroll
        for (int r = 0; r < 8; ++r) { pf[r] = 0.f; ph[r] = 0.f; pg[r] = 0.f; }

#pragma unroll
        for (int blk = 0; blk < 4; ++blk) {       // f-net hidden, 16 cols/blk
            v2f Bf; Bf.x = fB0[blk]; Bf.y = fB1[blk];
            v8f d = wmma_16x16x4(Af, Bf, fb1v[blk]);
#pragma unroll
            for (int r = 0; r < 8; ++r)
                pf[r] += softplus_f(d[r]) * fw2v[blk];
        }
#pragma unroll
        for (int blk = 0; blk < 4; ++blk) {       // h-net hidden
            v2f Bh; Bh.x = hB0[blk]; Bh.y = 0.f;
            v8f d = wmma_16x16x4(Ah, Bh, hb1v[blk]);
#pragma unroll
            for (int r = 0; r < 8; ++r)
                ph[r] += softplus_f(d[r]) * hw2v[blk];
        }
#pragma unroll
        for (int blk = 0; blk < 4; ++blk) {       // g-net hidden
            v2f Bg; Bg.x = gB0[blk]; Bg.y = 0.f;
            v8f d = wmma_16x16x4(Ah, Bg, gb1v[blk]);
#pragma unroll
            for (int r = 0; r < 8; ++r)
                pg[r] += softplus_f(d[r]) * gw2v[blk];
        }

        float fv = rowsum_select(pf, lane) + fb2s;
        float hv = rowsum_select(ph, lane) + hb2s;
        float gp = rowsum_select(pg, lane) + gb2s;

        // sigmoid without division: 1/gv == 1 + exp(-gp) exactly.
        float e    = __expf(-gp);
        float ginv = 1.f + e;                         // = 1 / sigmoid(gp)
        float gv   = __builtin_amdgcn_rcpf(ginv);     // sigmoid(gp)

        float u = (fv - hv) * ginv;
        acc += low ? (0.5f * u * u * dt) : 0.f;

        float dwv = dW[k * B_N + base + n16];
        z = fmaf(fv, dt, z) + gv * dwv * sdt;         // Euler-Maruyama
        if (!low) z = 0.f;
        if (low) zs[(k + 1) * B_N + base + lane] = z;
    }

    // per-wave sum of (kl + sum_k lr) over its 16 chains
    float am = low ? acc : 0.f;
    am += __shfl_xor(am, 1, 32);
    am += __shfl_xor(am, 2, 32);
    am += __shfl_xor(am, 4, 32);
    am += __shfl_xor(am, 8, 32);
    am += __shfl_xor(am, 16, 32);
    if (lane == 0) csums[blockIdx.x] = am;
}

// ---------------------------------------------------------------------------
// Kernel 3: projector + Gaussian log-likelihood, pointwise over (t,b),
// block-reduced into per-block partials (deterministic, no float atomics).
// ---------------------------------------------------------------------------
__global__ __launch_bounds__(256) void proj_loss_kernel(
    const float* __restrict__ xs,
    const float* __restrict__ zsbuf,
    const float* __restrict__ pw1, const float* __restrict__ pb1,
    const float* __restrict__ pw2, const float* __restrict__ pb2,
    const float* __restrict__ noise_std,
    float* __restrict__ dsums)
{
    int tid = threadIdx.x;
    int i = blockIdx.x * 256 + tid;
    float lp = 0.f;
    if (i < TB_N) {
        float z = zsbuf[i];
        float acc = pb2[0];
#pragma unroll 8
        for (int j = 0; j < H_N; ++j)
            acc += pw2[j] * softplus_f(fmaf(z, pw1[j], pb1[j]));
        float ns = noise_std[0];
        float d  = (xs[i] - acc) / ns;
        lp = -0.5f * d * d - logf(ns) - 0.91893853320467274f; // 0.5*log(2*pi)
    }
    __shared__ float sm[256];
    sm[tid] = lp;
    __syncthreads();
#pragma unroll
    for (int s = 128; s > 0; s >>= 1) {
        if (tid < s) sm[tid] += sm[tid + s];
        __syncthreads();
    }
    if (tid == 0) dsums[blockIdx.x] = sm[0];
}

// ---------------------------------------------------------------------------
// Kernel 4: deterministic finalize. out[0]=log_pxs, out[1]=logqp0+logqp_path.
// ---------------------------------------------------------------------------
__global__ __launch_bounds__(256) void finalize_kernel(
    const float* __restrict__ dsums, int nd,
    const float* __restrict__ csums, int nc,
    float* __restrict__ out)
{
    __shared__ float sm[256];
    int tid = threadIdx.x;

    float a = 0.f;
    for (int j = tid; j < nd; j += 256) a += dsums[j];
    sm[tid] = a; __syncthreads();
    for (int s = 128; s > 0; s >>= 1) { if (tid < s) sm[tid] += sm[tid + s]; __syncthreads(); }
    if (tid == 0) out[0] = sm[0] * (1.f / (float)B_N);
    __syncthreads();

    float b = 0.f;
    for (int j = tid; j < nc; j += 256) b += csums[j];
    sm[tid] = b; __syncthreads();
    for (int s = 128; s > 0; s >>= 1) { if (tid < s) sm[tid] += sm[tid + s]; __syncthreads(); }
    if (tid == 0) out[1] = sm[0] * (1.f / (float)B_N);
}

// ---------------------------------------------------------------------------
extern "C" void kernel_launch(void* const* d_in, const int* in_sizes, int n_in,
                              void* d_out, int out_size, void* d_ws, size_t ws_size,
                              hipStream_t stream)
{
    (void)in_sizes; (void)n_in; (void)out_size; (void)ws_size;
    const float* xs   = (const float*)d_in[0];
    const float* ts   = (const float*)d_in[1];
    const float* nstd = (const float*)d_in[2];
    const float* eps0 = (const float*)d_in[3];
    const float* dW   = (const float*)d_in[4];
    const float* ew1  = (const float*)d_in[5];
    const float* eb1  = (const float*)d_in[6];
    const float* ew2  = (const float*)d_in[7];
    const float* eb2  = (const float*)d_in[8];
    const float* qw   = (const float*)d_in[9];
    const float* qb   = (const float*)d_in[10];
    const float* fw1  = (const float*)d_in[11];
    const float* fb1  = (const float*)d_in[12];
    const float* fw2  = (const float*)d_in[13];
    const float* fb2  = (const float*)d_in[14];
    const float* hw1  = (const float*)d_in[15];
    const float* hb1  = (const float*)d_in[16];
    const float* hw2  = (const float*)d_in[17];
    const float* hb2  = (const float*)d_in[18];
    const float* gw1  = (const float*)d_in[19];
    const float* gb1  = (const float*)d_in[20];
    const float* gw2  = (const float*)d_in[21];
    const float* gb2  = (const float*)d_in[22];
    const float* pw1  = (const float*)d_in[23];
    const float* pb1  = (const float*)d_in[24];
    const float* pw2  = (const float*)d_in[25];
    const float* pb2  = (const float*)d_in[26];
    const float* pzm  = (const float*)d_in[27];
    const float* pzl  = (const float*)d_in[28];

    float* w     = (float*)d_ws;
    float* ctx   = w;                 // [T*B]
    float* zsbuf = w + TB_N;          // [T*B]
    float* csums = w + 2 * TB_N;      // [512]
    float* dsums = csums + 512;       // [4096]

    encoder_kernel<<<NBLK_PW, 256, 0, stream>>>(xs, ew1, eb1, ew2, eb2, ctx);
    sde_scan_kernel<<<B_N / 16, 32, 0, stream>>>(
        ts, eps0, dW, qw, qb,
        fw1, fb1, fw2, fb2, hw1, hb1, hw2, hb2,
        gw1, gb1, gw2, gb2, pzm, pzl, ctx, zsbuf, csums);
    proj_loss_kernel<<<NBLK_PW, 256, 0, stream>>>(xs, zsbuf, pw1, pb1, pw2, pb2, nstd, dsums);
    finalize_kernel<<<1, 256, 0, stream>>>(dsums, NBLK_PW, csums, B_N / 16, (float*)d_out);
}